// CoherentDONN_88450556494245
// MI455X (gfx1250) — compile-verified
//
#include <hip/hip_runtime.h>
#include <cmath>
#include <cstdint>

// ---------------------------------------------------------------------------
// Coherent DONN on MI455X (gfx1250, wave32).
//   rows_first : load x, *exp(i phi0), FFT512 rows
//   3x cols    : TDM tile load -> FFT512 cols, *H, IFFT512 cols   (in-place)
//   2x rows_mid: IFFT512 rows (/512), *exp(i phi_l), FFT512 rows  (in-place)
//   rows_last  : IFFT512 rows (/512), intensity -> .x in place
//   gemm(WMMA) : (128 x 262144) @ fc_w^T, v_wmma_f32_16x16x4_f32, TDM-staged
//                double-buffered tiles, K split 128-way, deterministic reduce.
// ---------------------------------------------------------------------------

#define RESN   512
#define BATCH  128
#define NCLS   10
#define IMG    (RESN * RESN)          // 262144
#define RTILE  4                      // rows per 256-thread block
#define CTILE  8                      // cols per 512-thread block
#define CSTRIDE 9                     // LDS stride (float2) for col tile (bank pad)
#define KSPLIT 128
#define KCHUNK (IMG / KSPLIT)         // 2048 K per block
#define KSTEP  128                    // K per staged tile
#define MTILES (BATCH / 16)           // 8

#define TWO_PI_D 6.283185307179586476925286766559
#define LAMBDA_D 5.32e-07
#define ZDIST_D  0.035
#define KZ_D     (TWO_PI_D * ZDIST_D / LAMBDA_D)
#define PLZ_D    (3.1415926535897932384626433832795 * LAMBDA_D * ZDIST_D)
#define FREQ_SCALE 1953.125f          // 1/(512 * 1e-6)
#define INV_RES    (1.0f / 512.0f)

typedef __attribute__((ext_vector_type(2))) float v2f;
typedef __attribute__((ext_vector_type(8))) float v8f;

#if __has_builtin(__builtin_amdgcn_tensor_load_to_lds) && \
    __has_builtin(__builtin_amdgcn_s_wait_tensorcnt)
#define HAVE_TDM 1
#else
#define HAVE_TDM 0
#endif

#if HAVE_TDM
typedef unsigned int u32x4 __attribute__((ext_vector_type(4)));
typedef int i32x8 __attribute__((ext_vector_type(8)));
typedef int i32x4 __attribute__((ext_vector_type(4)));

// Issue one 2D TDM tile load: tile_d1 rows of tile_d0 elements, row stride
// stride0 (elements), into LDS at lds_off (bytes). w0flags carries
// data_size / pad_enable / pad_interval / pad_amount per D# group-1 word 0.
__device__ __forceinline__ void tdm_load_2d(unsigned lds_off, const void* gptr,
                                            unsigned w0flags,
                                            unsigned tensor_d0, unsigned tensor_d1,
                                            unsigned tile_d0, unsigned tile_d1,
                                            unsigned long long stride0) {
  unsigned long long ga = (unsigned long long)(uintptr_t)gptr;
  u32x4 g0;
  g0.x = 1u;                                               // count=1 (valid D#)
  g0.y = lds_off;                                          // lds_addr
  g0.z = (unsigned)(ga & 0xffffffffull);                   // global_addr[31:0]
  g0.w = (unsigned)((ga >> 32) & 0x1ffffffull) | 0x80000000u; // [56:32] | type=2
  i32x8 g1;
  g1[0] = (int)w0flags;                                    // mask=0 | ds | pad
  g1[1] = (int)((tensor_d0 & 0xffffu) << 16);              // dim0[15:0]
  g1[2] = (int)(((tensor_d0 >> 16) & 0xffffu) | ((tensor_d1 & 0xffffu) << 16));
  g1[3] = (int)(((tensor_d1 >> 16) & 0xffffu) | ((tile_d0 & 0xffffu) << 16));
  g1[4] = (int)(tile_d1 & 0xffffu);                        // tile_d1 | tile_d2=0
  g1[5] = (int)(stride0 & 0xffffffffull);                  // dim0_stride[31:0]
  g1[6] = (int)((stride0 >> 32) & 0xffffull);              // dim0_stride[47:32]
  g1[7] = 0;
  i32x4 z4 = {0, 0, 0, 0};
  i32x8 z8 = {0, 0, 0, 0, 0, 0, 0, 0};
  // 6-arg variant (amdgpu-toolchain / clang-23): (g0, g1, g2, g3, g4, cpol)
  __builtin_amdgcn_tensor_load_to_lds(g0, g1, z4, z4, z8, 0);
}
// data_size codes: 2 = 4B, 3 = 8B
#define TDM_FLAGS_A ((3u << 16) | (1u << 20) | (7u << 22) | (1u << 25)) // 8B, pad 2dw/256dw
#define TDM_FLAGS_B ((2u << 16) | (1u << 20) | (6u << 22) | (0u << 25)) // 4B, pad 1dw/128dw
#define TDM_FLAGS_C ((3u << 16) | (1u << 20) | (3u << 22) | (1u << 25)) // 8B, pad 2dw/16dw
#endif

__device__ __forceinline__ float2 cmulf(float2 a, float2 b) {
  return make_float2(a.x * b.x - a.y * b.y, a.x * b.y + a.y * b.x);
}

template <bool INV>
__device__ __forceinline__ float2 rot90(float2 a) {   // *(-i) fwd, *(+i) inv
  return INV ? make_float2(-a.y, a.x) : make_float2(a.y, -a.x);
}

// 8-point DIF FFT; output bit-reversed: X[k] = v[brev3(k)].
template <bool INV>
__device__ __forceinline__ void fft8(float2 (&v)[8]) {
  const float S = 0.70710678118654752440f;
  const float2 w1 = INV ? make_float2(S, S) : make_float2(S, -S);
  const float2 w3 = INV ? make_float2(-S, S) : make_float2(-S, -S);
  float2 t;
#define BFLY(a, b)                                              \
  t = a;                                                        \
  a = make_float2(t.x + b.x, t.y + b.y);                        \
  b = make_float2(t.x - b.x, t.y - b.y);
  BFLY(v[0], v[4]); BFLY(v[1], v[5]); BFLY(v[2], v[6]); BFLY(v[3], v[7]);
  v[5] = cmulf(v[5], w1);
  v[6] = rot90<INV>(v[6]);
  v[7] = cmulf(v[7], w3);
  BFLY(v[0], v[2]); BFLY(v[1], v[3]);
  v[3] = rot90<INV>(v[3]);
  BFLY(v[4], v[6]); BFLY(v[5], v[7]);
  v[7] = rot90<INV>(v[7]);
  BFLY(v[0], v[1]); BFLY(v[2], v[3]); BFLY(v[4], v[5]); BFLY(v[6], v[7]);
#undef BFLY
}

// Stockham radix-8^3 512-pt FFT, 64 threads per transform.
// In/out: v[r] = data[j + 64r] (natural order). LDS element i at buf[i*STRIDE].
template <bool INV, int STRIDE>
__device__ __forceinline__ void fft512_ldst(float2* buf, float2 (&v)[8], int j) {
  const float SGN = INV ? 1.0f : -1.0f;
  const int brev[8] = {0, 4, 2, 6, 1, 5, 3, 7};
  fft8<INV>(v);                                             // stage 1 (Ns=1)
  __syncthreads();
#pragma unroll
  for (int r = 0; r < 8; ++r) buf[(8 * j + r) * STRIDE] = v[brev[r]];
  __syncthreads();
#pragma unroll
  for (int r = 0; r < 8; ++r) v[r] = buf[(j + 64 * r) * STRIDE];
  {
    float a = SGN * 0.09817477042468103870f * (float)(j & 7);  // 2pi/64
#pragma unroll
    for (int r = 1; r < 8; ++r) {
      float s, c;
      __sincosf(a * (float)r, &s, &c);
      v[r] = cmulf(v[r], make_float2(c, s));
    }
  }
  fft8<INV>(v);                                             // stage 2 (Ns=8)
  __syncthreads();
  {
    int d2 = ((j >> 3) << 6) + (j & 7);
#pragma unroll
    for (int r = 0; r < 8; ++r) buf[(d2 + 8 * r) * STRIDE] = v[brev[r]];
  }
  __syncthreads();
#pragma unroll
  for (int r = 0; r < 8; ++r) v[r] = buf[(j + 64 * r) * STRIDE];
  {
    float a = SGN * 0.01227184630308512983f * (float)j;        // 2pi/512
#pragma unroll
    for (int r = 1; r < 8; ++r) {
      float s, c;
      __sincosf(a * (float)r, &s, &c);
      v[r] = cmulf(v[r], make_float2(c, s));
    }
  }
  fft8<INV>(v);                                             // stage 3 (Ns=64)
  float2 w[8];
#pragma unroll
  for (int r = 0; r < 8; ++r) w[r] = v[brev[r]];
#pragma unroll
  for (int r = 0; r < 8; ++r) v[r] = w[r];
}

// ---------------- H(ky,kx) precompute (1/512 IFFT-col scale folded in) -----
__global__ void donn_make_h(float2* __restrict__ H) {
  int i = blockIdx.x * 256 + threadIdx.x;
  int ky = i >> 9, kx = i & 511;
  float fy = (float)(ky < 256 ? ky : ky - 512) * FREQ_SCALE;
  float fx = (float)(kx < 256 ? kx : kx - 512) * FREQ_SCALE;
  double ph = KZ_D - PLZ_D * ((double)fx * (double)fx + (double)fy * (double)fy);
  ph -= TWO_PI_D * floor(ph / TWO_PI_D);
  float s, c;
  __sincosf((float)ph, &s, &c);
  H[i] = make_float2(c * INV_RES, s * INV_RES);
}

// ---------------- row passes ------------------------------------------------
__global__ __launch_bounds__(256) void donn_rows_first(
    const float* __restrict__ x, const float* __restrict__ ph0,
    float2* __restrict__ field) {
  __shared__ float2 lds[RTILE * RESN];
  const int g = threadIdx.x >> 6, j = threadIdx.x & 63;
  const int row = blockIdx.x * RTILE + g;
  const int b = row >> 9, y = row & 511;
  const size_t base = (size_t)b * IMG + (size_t)y * RESN;
  float2 v[8];
#pragma unroll
  for (int r = 0; r < 8; ++r) {
    int xc = j + 64 * r;
    float xr = x[base + xc];
    float p = ph0[(size_t)y * RESN + xc];
    float s, c;
    __sincosf(p, &s, &c);
    v[r] = make_float2(xr * c, xr * s);
  }
  fft512_ldst<false, 1>(lds + g * RESN, v, j);
#pragma unroll
  for (int r = 0; r < 8; ++r) field[base + j + 64 * r] = v[r];
}

__global__ __launch_bounds__(256) void donn_rows_mid(
    float2* __restrict__ field, const float* __restrict__ ph) {
  __shared__ float2 lds[RTILE * RESN];
  const int g = threadIdx.x >> 6, j = threadIdx.x & 63;
  const int row = blockIdx.x * RTILE + g;
  const int b = row >> 9, y = row & 511;
  const size_t base = (size_t)b * IMG + (size_t)y * RESN;
  float2 v[8];
#pragma unroll
  for (int r = 0; r < 8; ++r) {
    float2 f = field[base + j + 64 * r];
    v[r] = make_float2(f.x * INV_RES, f.y * INV_RES);   // row-IFFT 1/512
  }
  fft512_ldst<true, 1>(lds + g * RESN, v, j);           // IFFT rows
#pragma unroll
  for (int r = 0; r < 8; ++r) {                         // *exp(i phi_l)
    int xc = j + 64 * r;
    float p = ph[(size_t)y * RESN + xc];
    float s, c;
    __sincosf(p, &s, &c);
    v[r] = cmulf(v[r], make_float2(c, s));
  }
  fft512_ldst<false, 1>(lds + g * RESN, v, j);          // FFT rows
#pragma unroll
  for (int r = 0; r < 8; ++r) field[base + j + 64 * r] = v[r];
}

__global__ __launch_bounds__(256) void donn_rows_last(float2* __restrict__ field) {
  __shared__ float2 lds[RTILE * RESN];
  const int g = threadIdx.x >> 6, j = threadIdx.x & 63;
  const int row = blockIdx.x * RTILE + g;
  const int b = row >> 9, y = row & 511;
  const size_t base = (size_t)b * IMG + (size_t)y * RESN;
  float2 v[8];
#pragma unroll
  for (int r = 0; r < 8; ++r) {
    float2 f = field[base + j + 64 * r];
    v[r] = make_float2(f.x * INV_RES, f.y * INV_RES);
  }
  fft512_ldst<true, 1>(lds + g * RESN, v, j);
#pragma unroll
  for (int r = 0; r < 8; ++r)                           // intensity, in place
    field[base + j + 64 * r] =
        make_float2(v[r].x * v[r].x + v[r].y * v[r].y, 0.0f);
}

// ---------------- column pass: FFTy * H * IFFTy, LDS/reg resident ----------
__global__ __launch_bounds__(512) void donn_cols(float2* __restrict__ field,
                                                 const float2* __restrict__ H) {
  __shared__ float2 tile[RESN * CSTRIDE];               // 36864 B, stride-9 rows
  const int t = threadIdx.x;
  const int g = t >> 6, j = t & 63;                     // column-in-tile, FFT lane
  const int b = blockIdx.x / (RESN / CTILE);
  const int x0 = (blockIdx.x % (RESN / CTILE)) * CTILE;
  const size_t base = (size_t)b * IMG;
#if HAVE_TDM
  if (t < 32) {                                         // wave 0 drives the DMA
    tdm_load_2d((unsigned)(size_t)(void*)tile, field + base + x0, TDM_FLAGS_C,
                /*tensor_d0=*/RESN, /*tensor_d1=*/RESN,
                /*tile_d0=*/CTILE, /*tile_d1=*/RESN, /*stride0=*/RESN);
    __builtin_amdgcn_s_wait_tensorcnt((short)0);
  }
#else
#pragma unroll
  for (int i = 0; i < 8; ++i) {                         // coalesced tile load
    int idx = t + i * 512;
    int y = idx >> 3, c = idx & 7;
    tile[y * CSTRIDE + c] = field[base + (size_t)y * RESN + x0 + c];
  }
#endif
  __syncthreads();
  float2 v[8];
#pragma unroll
  for (int r = 0; r < 8; ++r) v[r] = tile[(j + 64 * r) * CSTRIDE + g];
  fft512_ldst<false, CSTRIDE>(tile + g, v, j);          // FFT cols
#pragma unroll
  for (int r = 0; r < 8; ++r) {                         // *H (H has 1/512)
    int ky = j + 64 * r;
    v[r] = cmulf(v[r], H[(size_t)ky * RESN + x0 + g]);
  }
  fft512_ldst<true, CSTRIDE>(tile + g, v, j);           // IFFT cols
  __syncthreads();
#pragma unroll
  for (int r = 0; r < 8; ++r) tile[(j + 64 * r) * CSTRIDE + g] = v[r];
  __syncthreads();
#pragma unroll
  for (int i = 0; i < 8; ++i) {                         // coalesced store
    int idx = t + i * 512;
    int y = idx >> 3, c = idx & 7;
    field[base + (size_t)y * RESN + x0 + c] = tile[y * CSTRIDE + c];
  }
}

// ---------------- final FC via WMMA f32 16x16x4, TDM double-buffered -------
__global__ __launch_bounds__(256) void donn_gemm(const float2* __restrict__ intens,
                                                 const float* __restrict__ fc_w,
                                                 float* __restrict__ partial) {
  __shared__ float2 At[2][16 * 129];                    // TDM-padded rows (129)
  __shared__ float  Bt[2][16 * 129];
  __shared__ float  red[8 * 256];
  const int t = threadIdx.x;
  const int wave = t >> 5, lane = t & 31;
  const int mt = blockIdx.x / KSPLIT;
  const int ks = blockIdx.x % KSPLIT;
  const size_t k0 = (size_t)ks * KCHUNK;
  const int nch = KCHUNK / KSTEP;                       // 16
  v8f acc = {};

#if HAVE_TDM
  if (wave == 0) {                                      // prologue: chunk 0
    tdm_load_2d((unsigned)(size_t)(void*)&At[0][0],
                intens + (size_t)(mt * 16) * IMG + k0, TDM_FLAGS_A,
                IMG, 16, KSTEP, 16, IMG);
    tdm_load_2d((unsigned)(size_t)(void*)&Bt[0][0],
                fc_w + k0, TDM_FLAGS_B,
                IMG, NCLS /*rows >=10 zero-fill*/, KSTEP, 16, IMG);
  }
#endif
  for (int c = 0; c < nch; ++c) {
    const int cur = c & 1;
#if HAVE_TDM
    if (wave == 0) {
      if (c + 1 < nch) {                                // overlap DMA of c+1
        const int nxt = (c + 1) & 1;
        tdm_load_2d((unsigned)(size_t)(void*)&At[nxt][0],
                    intens + (size_t)(mt * 16) * IMG + k0 + (c + 1) * KSTEP,
                    TDM_FLAGS_A, IMG, 16, KSTEP, 16, IMG);
        tdm_load_2d((unsigned)(size_t)(void*)&Bt[nxt][0],
                    fc_w + k0 + (c + 1) * KSTEP,
                    TDM_FLAGS_B, IMG, NCLS, KSTEP, 16, IMG);
        __builtin_amdgcn_s_wait_tensorcnt((short)2);    // chunk c landed
      } else {
        __builtin_amdgcn_s_wait_tensorcnt((short)0);
      }
    }
    __syncthreads();
#else
    __syncthreads();
    for (int idx = t; idx < 16 * KSTEP; idx += 256) {
      int m = idx >> 7, k = idx & (KSTEP - 1);
      At[cur][m * 129 + k] =
          intens[(size_t)(mt * 16 + m) * IMG + k0 + c * KSTEP + k];
      Bt[cur][m * 129 + k] =
          (m < NCLS) ? fc_w[(size_t)m * IMG + k0 + c * KSTEP + k] : 0.0f;
    }
    if (c + 1 < nch)
      __builtin_prefetch(&intens[(size_t)(mt * 16) * IMG + k0 + (c + 1) * KSTEP + t], 0, 1);
    __syncthreads();
#endif
    const int kb = wave * 16;                           // 16 K-values per wave
    const int half2 = (lane >> 4) * 2;                  // lanes 16-31: K+2
    const int mr = lane & 15;
#pragma unroll
    for (int kk = 0; kk < 16; kk += 4) {
      int kidx = kb + kk + half2;
      v2f a, bb;
      a.x = At[cur][mr * 129 + kidx].x;
      a.y = At[cur][mr * 129 + kidx + 1].x;
      bb.x = Bt[cur][mr * 129 + kidx];
      bb.y = Bt[cur][mr * 129 + kidx + 1];
      acc = __builtin_amdgcn_wmma_f32_16x16x4_f32(false, a, false, bb,
                                                  (short)0, acc, false, false);
    }
    __syncthreads();                                    // reads done before next DMA
  }
#pragma unroll
  for (int r = 0; r < 8; ++r)                           // C: M=r(+8), N=lane%16
    red[wave * 256 + (r + (lane >> 4) * 8) * 16 + (lane & 15)] = acc[r];
  __syncthreads();
  float s = 0.0f;
#pragma unroll
  for (int w = 0; w < 8; ++w) s += red[w * 256 + t];
  partial[((size_t)mt * KSPLIT + ks) * 256 + t] = s;
}

__global__ void donn_reduce(const float* __restrict__ partial,
                            const float* __restrict__ fc_b,
                            float* __restrict__ out) {
  int i = blockIdx.x * 256 + threadIdx.x;
  if (i >= BATCH * NCLS) return;
  int m = i / NCLS, n = i % NCLS;
  int mt = m >> 4, ml = m & 15;
  float s = fc_b[n];
  for (int ks = 0; ks < KSPLIT; ++ks)
    s += partial[((size_t)mt * KSPLIT + ks) * 256 + ml * 16 + n];
  out[i] = s;
}

// ---------------------------------------------------------------------------
extern "C" void kernel_launch(void* const* d_in, const int* in_sizes, int n_in,
                              void* d_out, int out_size, void* d_ws, size_t ws_size,
                              hipStream_t stream) {
  (void)in_sizes; (void)n_in; (void)out_size; (void)ws_size;
  const float* x      = (const float*)d_in[0];
  const float* phases = (const float*)d_in[1];
  const float* fc_w   = (const float*)d_in[2];
  const float* fc_b   = (const float*)d_in[3];
  float* out = (float*)d_out;

  char* ws = (char*)d_ws;
  float2* field   = (float2*)ws;                               // 256 MB
  float*  partial = (float*)(ws + 268435456ULL);               // 1 MB
  float2* H       = (float2*)(ws + 268435456ULL + 1048576ULL); // 2 MB

  donn_make_h<<<IMG / 256, 256, 0, stream>>>(H);
  donn_rows_first<<<BATCH * RESN / RTILE, 256, 0, stream>>>(x, phases, field);
  for (int l = 0; l < 3; ++l) {
    donn_cols<<<BATCH * (RESN / CTILE), 512, 0, stream>>>(field, H);
    if (l < 2)
      donn_rows_mid<<<BATCH * RESN / RTILE, 256, 0, stream>>>(
          field, phases + (size_t)(l + 1) * IMG);
  }
  donn_rows_last<<<BATCH * RESN / RTILE, 256, 0, stream>>>(field);
  donn_gemm<<<MTILES * KSPLIT, 256, 0, stream>>>(field, fc_w, partial);
  donn_reduce<<<(BATCH * NCLS + 255) / 256, 256, 0, stream>>>(partial, fc_b, out);
}